// PCA_45612552684148
// MI455X (gfx1250) — compile-verified
//
#include <hip/hip_runtime.h>
#include <hip/hip_bf16.h>
#include <math.h>

// ---------------------------------------------------------------------------
// Problem: x [B=8, C=3, T=512, H=64, W=64] f32
//   xm = mean(x, axis=(3,4))                       -> [8,3,512]
//   per-batch PCA on [3,512]: out[b] = Vt[1] * explained_variance[1]  -> [8,512]
// Stage 1 is memory-bound (201 MB @ 23.3 TB/s ~ 8.6us): streaming reduction.
// Stage 2 uses V_WMMA_F32_16X16X4_F32 to form the Gram matrix G = Xc*Xc^T.
// ---------------------------------------------------------------------------

typedef __attribute__((ext_vector_type(2))) float v2f;
typedef __attribute__((ext_vector_type(4))) float f4;
typedef __attribute__((ext_vector_type(8))) float v8f;

#define B_DIM  8
#define C_DIM  3
#define T_DIM  512
#define HW_DIM 4096   // 64*64
#define N_ROWS (B_DIM * C_DIM * T_DIM)  // 12288
#define XC_STRIDE 516 // breaks 64-bank conflicts of stride-512; keeps 8B align

// ---------------------------------------------------------------------------
// Stage 1: one workgroup (256 threads = 8 waves) per (b,c,t) row of 4096 f32.
// Non-temporal float4 loads (single-pass stream, keep L2 clean).
// ---------------------------------------------------------------------------
__global__ __launch_bounds__(256) void pca_rowmean_kernel(
    const float* __restrict__ x, float* __restrict__ xm) {
  const int row = blockIdx.x;                       // (b*3 + c)*512 + t
  const f4* p = (const f4*)(x + (size_t)row * HW_DIM);
  const int tid = threadIdx.x;

  float s = 0.0f;
#pragma unroll
  for (int i = 0; i < 4; ++i) {                     // 4 x float4 = 16 floats/thread
    f4 v = __builtin_nontemporal_load(&p[tid + i * 256]);
    s += (v.x + v.y) + (v.z + v.w);
  }
  // wave32 tree reduce
#pragma unroll
  for (int off = 16; off >= 1; off >>= 1) s += __shfl_xor(s, off, 32);

  __shared__ float ws[8];
  const int wave = tid >> 5, lane = tid & 31;
  if (lane == 0) ws[wave] = s;
  __syncthreads();
  if (tid == 0) {
    float t = 0.0f;
#pragma unroll
    for (int w = 0; w < 8; ++w) t += ws[w];
    xm[row] = t * (1.0f / (float)HW_DIM);
  }
}

__device__ __forceinline__ void cross3(float ax, float ay, float az,
                                       float bx, float by, float bz,
                                       float& cx, float& cy, float& cz) {
  cx = ay * bz - az * by;
  cy = az * bx - ax * bz;
  cz = ax * by - ay * bx;
}

// ---------------------------------------------------------------------------
// Stage 2: one wave per batch.
//   1) center columns of xm[b] -> Xc rows 0..2 in LDS; rows 3..15 zero-padded
//      so fragment loads are unconditional (no EXEC-masked ds loads) and
//      vectorize to ds_load_b64.
//   2) G = Xc * Xc^T via 128 x v_wmma_f32_16x16x4_f32 (B fragment == A fragment
//      since B = A^T under the mirrored f32 A/B VGPR layouts)
//   3) closed-form symmetric 3x3 eigen: lambda_1 = middle eigenvalue
//      (columns of Xc sum to 0 -> rank<=2 -> smallest eigenvalue ~ 0)
//   4) eigenvector via cross products of rows of (G - lambda_1*I),
//      sklearn svd_flip sign (first argmax of |u| made positive, sign(0)->+1)
//   5) out[b][t] = sgn * (u . Xc[:,t]) * sqrt(lambda_1) / 2
// ---------------------------------------------------------------------------
__global__ __launch_bounds__(32) void pca_solve_kernel(
    const float* __restrict__ xm, float* __restrict__ out) {
  const int b = blockIdx.x;
  const int lane = threadIdx.x;

  __shared__ float Xc[16][XC_STRIDE];   // rows 3..15 are zero padding
  __shared__ float Gs[3][3];

  // zero the padding rows (only once; 13*516 floats)
  {
    float* pad = &Xc[3][0];
    for (int i = lane; i < 13 * XC_STRIDE; i += 32) pad[i] = 0.0f;
  }

  const float* src = xm + b * (C_DIM * T_DIM);
  for (int t = lane; t < T_DIM; t += 32) {
    const float x0 = src[t];
    const float x1 = src[T_DIM + t];
    const float x2 = src[2 * T_DIM + t];
    const float m = (x0 + x1 + x2) * (1.0f / 3.0f);
    Xc[0][t] = x0 - m;
    Xc[1][t] = x1 - m;
    Xc[2][t] = x2 - m;
  }
  __syncthreads();

  // --- Gram matrix via WMMA f32 16x16x4, K-loop over 512 ---
  // f32 A 16x4 layout: lanes 0-15 -> M=lane, K={k0,k0+1}; lanes 16-31 -> K={k0+2,k0+3}
  // f32 B 4x16 mirrors it, so for B = A^T the fragments are identical.
  const int half = lane >> 4;       // selects K pair
  const int lo = lane & 15;         // M (for A) == N (for B)
  const float* rowp = &Xc[lo][2 * half];  // 8B-aligned for even k0

  v8f acc = {};
#pragma unroll 4
  for (int k0 = 0; k0 < T_DIM; k0 += 4) {
    const v2f a = *(const v2f*)(rowp + k0);   // ds_load_b64, bank-conflict-free
    acc = __builtin_amdgcn_wmma_f32_16x16x4_f32(
        /*neg_a=*/false, a, /*neg_b=*/false, a,
        /*c_mod=*/(short)0, acc, /*reuse_a=*/false, /*reuse_b=*/false);
  }

  // D layout: VGPR i, lane j (j<16) = D[i][j] -> G[i][j] for i,j < 3
  if (lane < 3) {
    Gs[0][lane] = acc[0];
    Gs[1][lane] = acc[1];
    Gs[2][lane] = acc[2];
  }
  __syncthreads();

  const float g00 = Gs[0][0], g01 = Gs[0][1], g02 = Gs[0][2];
  const float g11 = Gs[1][1], g12 = Gs[1][2], g22 = Gs[2][2];

  // --- closed-form eigenvalues of symmetric 3x3 (trigonometric cubic) ---
  const float q = (g00 + g11 + g22) * (1.0f / 3.0f);
  const float a00 = g00 - q, a11 = g11 - q, a22 = g22 - q;
  const float p1 = g01 * g01 + g02 * g02 + g12 * g12;
  const float p2 = a00 * a00 + a11 * a11 + a22 * a22 + 2.0f * p1;
  const float p = sqrtf(p2 * (1.0f / 6.0f));
  const float invp = (p > 1e-30f) ? (1.0f / p) : 0.0f;
  const float b00 = a00 * invp, b11 = a11 * invp, b22 = a22 * invp;
  const float b01 = g01 * invp, b02 = g02 * invp, b12 = g12 * invp;
  float detB = b00 * (b11 * b22 - b12 * b12)
             - b01 * (b01 * b22 - b12 * b02)
             + b02 * (b01 * b12 - b11 * b02);
  float r = 0.5f * detB;
  r = fminf(1.0f, fmaxf(-1.0f, r));
  const float phi = acosf(r) * (1.0f / 3.0f);
  const float e0 = q + 2.0f * p * cosf(phi);                       // largest
  const float e2 = q + 2.0f * p * cosf(phi + 2.0943951023931953f); // smallest
  const float e1 = 3.0f * q - e0 - e2;                             // middle = lambda_1

  // --- eigenvector for e1: best cross product of rows of (G - e1*I) ---
  const float m00 = g00 - e1, m11 = g11 - e1, m22 = g22 - e1;
  float c0x, c0y, c0z, c1x, c1y, c1z, c2x, c2y, c2z;
  cross3(m00, g01, g02, g01, m11, g12, c0x, c0y, c0z);
  cross3(m00, g01, g02, g02, g12, m22, c1x, c1y, c1z);
  cross3(g01, m11, g12, g02, g12, m22, c2x, c2y, c2z);
  const float n0 = c0x * c0x + c0y * c0y + c0z * c0z;
  const float n1 = c1x * c1x + c1y * c1y + c1z * c1z;
  const float n2 = c2x * c2x + c2y * c2y + c2z * c2z;
  float ux = c0x, uy = c0y, uz = c0z, nn = n0;
  if (n1 > nn) { ux = c1x; uy = c1y; uz = c1z; nn = n1; }
  if (n2 > nn) { ux = c2x; uy = c2y; uz = c2z; nn = n2; }
  if (nn < 1e-30f) { ux = 1.0f; uy = 0.0f; uz = 0.0f; nn = 1.0f; }
  const float inv_n = rsqrtf(nn);
  ux *= inv_n; uy *= inv_n; uz *= inv_n;

  // svd_flip sign: first argmax of |u|, sign(0) -> +1
  const float au0 = fabsf(ux), au1 = fabsf(uy), au2 = fabsf(uz);
  float picked = ux, av = au0;
  if (au1 > av) { picked = uy; av = au1; }
  if (au2 > av) { picked = uz; }
  const float sgn = (picked < 0.0f) ? -1.0f : 1.0f;

  const float S1 = sqrtf(fmaxf(e1, 0.0f));
  const float scale = sgn * 0.5f * S1;   // Vt[1]*ev[1] = (u^T Xc / S1) * S1^2/2

  float* o = out + b * T_DIM;
  for (int t = lane; t < T_DIM; t += 32) {
    o[t] = (ux * Xc[0][t] + uy * Xc[1][t] + uz * Xc[2][t]) * scale;
  }
}

extern "C" void kernel_launch(void* const* d_in, const int* in_sizes, int n_in,
                              void* d_out, int out_size, void* d_ws, size_t ws_size,
                              hipStream_t stream) {
  (void)in_sizes; (void)n_in; (void)out_size; (void)ws_size;
  const float* x = (const float*)d_in[0];
  float* xm = (float*)d_ws;        // 12288 f32 scratch
  float* out = (float*)d_out;      // 8*512 f32

  pca_rowmean_kernel<<<N_ROWS, 256, 0, stream>>>(x, xm);
  pca_solve_kernel<<<B_DIM, 32, 0, stream>>>(xm, out);
}